// BlendHydroV1_18854906429629
// MI455X (gfx1250) — compile-verified
//
#include <hip/hip_runtime.h>
#include <stdint.h>

// Problem constants (from reference)
#define S_LEN   730
#define G_N     4000
#define LENF    15
#define NEARZ   1e-5f
#define RAWSTR  38          // 18*NMUL + 2 floats per gauge

// Scan-kernel tiling: small blocks to spread the latency-bound scan over many WGPs.
#define GB      32          // gauges per block  (4000/32 = 125 blocks, no remainder)
#define GB3     (GB * 3)    // floats per timestep row in LDS (96)
#define NTHR    (GB * 2)    // 64 threads = 2 waves per block
#define CHUNK   16          // timesteps staged per LDS buffer
#define RPW     (CHUNK / 2) // rows per wave (2 waves per block)

__device__ __constant__ float cHBV_LB[12] = {1.0f, 50.0f, 0.05f, 0.01f, 0.001f, 0.2f, 0.0f, 0.0f, -2.5f, 0.5f, 0.0f, 0.0f};
__device__ __constant__ float cHBV_UB[12] = {6.0f, 1000.0f, 0.9f, 0.5f, 0.2f, 1.0f, 10.0f, 100.0f, 2.5f, 10.0f, 0.1f, 0.2f};
__device__ __constant__ float cEXP_LB[6]  = {0.0f, 100.0f, 10.0f, 0.0f, 0.0f, -3.0f};
__device__ __constant__ float cEXP_UB[6]  = {0.1f, 1500.0f, 50.0f, 5.0f, 3.0f, 0.0f};

typedef unsigned int u32x4 __attribute__((ext_vector_type(4)));
typedef int          i32x4 __attribute__((ext_vector_type(4)));
typedef int          i32x8 __attribute__((ext_vector_type(8)));

// Issue one TDM 2D tile load: nrows timesteps x GB3 contiguous f32 (row stride 3*G_N
// elements) into LDS at lds_byte_off.
__device__ __forceinline__ void tdm_issue(uint64_t ga, uint32_t lds_byte_off, int nrows) {
  u32x4 d0;
  d0[0] = 1u;                                   // count=1, user descriptor, no gather
  d0[1] = lds_byte_off;                         // lds_addr (bytes)
  d0[2] = (uint32_t)ga;                         // global_addr[31:0]
  d0[3] = (uint32_t)((ga >> 32) & 0x1FFFFFFu)   // global_addr[56:32]
          | (2u << 30);                         // type = 2 ("image")
  const uint32_t dim0 = 3u * (uint32_t)G_N;     // tensor row length (elements)
  i32x8 d1;
  d1[0] = (int)(2u << 16);                      // workgroup_mask=0, data_size=4B
  d1[1] = (int)((dim0 & 0xFFFFu) << 16);        // tensor_dim0[15:0]
  d1[2] = (int)(((dim0 >> 16) & 0xFFFFu) | (((uint32_t)S_LEN & 0xFFFFu) << 16));
  d1[3] = (int)((((uint32_t)S_LEN >> 16) & 0xFFFFu) | (((uint32_t)GB3 & 0xFFFFu) << 16)); // tile_dim0 = GB3
  d1[4] = (int)((uint32_t)nrows & 0xFFFFu);     // tile_dim1 = rows; tile_dim2 = 0 (2D)
  d1[5] = (int)dim0;                            // tensor_dim0_stride[31:0]
  d1[6] = 0;
  d1[7] = 0;
  i32x4 z4 = {0, 0, 0, 0};
#if defined(__clang_major__) && (__clang_major__ >= 23)
  i32x8 z8 = {0, 0, 0, 0, 0, 0, 0, 0};
  __builtin_amdgcn_tensor_load_to_lds(d0, d1, z4, z4, z8, 0);
#else
  __builtin_amdgcn_tensor_load_to_lds(d0, d1, z4, z4, 0);
#endif
}

// Kernel 1: fused HBV + ExpHydro sequential scan.
// One thread per (gauge, mul) pair; pair-adjacent lanes so the NMUL mean is a
// lane shuffle; TDM double-buffered input stream, one chunk of lookahead.
extern "C" __global__ void __launch_bounds__(NTHR)
hydro_scan_kernel(const float* __restrict__ x, const float* __restrict__ raw,
                  float* __restrict__ qsim) {
  __shared__ float xs[2 * CHUNK * GB3];   // 12 KB double buffer

  const int tid  = threadIdx.x;
  const int wave = tid >> 5;              // 0..1
  const int g0   = blockIdx.x * GB;
  const int pl   = tid >> 1;              // local gauge 0..31
  const int m    = tid & 1;               // NMUL index
  const int g    = g0 + pl;               // always < G_N (4000 % 32 == 0)

  // Descale parameters (reference reshape: raw[g, i*2+m])
  float hp[12], ep[6];
#pragma unroll
  for (int i = 0; i < 12; ++i) {
    float r = raw[g * RAWSTR + i * 2 + m];
    hp[i] = cHBV_LB[i] + r * (cHBV_UB[i] - cHBV_LB[i]);
  }
#pragma unroll
  for (int i = 0; i < 6; ++i) {
    float r = raw[g * RAWSTR + (12 + i) * 2 + m];
    ep[i] = cEXP_LB[i] + r * (cEXP_UB[i] - cEXP_LB[i]);
  }
  const float beta = hp[0], fc = hp[1], k0 = hp[2], k1 = hp[3], k2 = hp[4], lp = hp[5];
  const float perc = hp[6], uzl = hp[7], ttp = hp[8], cfmax = hp[9], cfr = hp[10], cwh = hp[11];
  const float f = ep[0], smax = ep[1], qmax = ep[2], df = ep[3], tmax = ep[4], tmin = ep[5];
  const float inv_fc = 1.0f / fc, inv_lpfc = 1.0f / (lp * fc), inv_smax = 1.0f / smax;

  // States
  float sp = NEARZ, mw = NEARZ, sm = 0.5f * fc, suz = NEARZ, slz = NEARZ;  // HBV
  float s0 = NEARZ, s1 = 0.5f * smax;                                      // ExpHydro

  const uint32_t lds_base = (uint32_t)(uintptr_t)&xs[0];
  const uint64_t rowbytes = (uint64_t)(3 * G_N) * 4ull;   // global row stride (bytes)
  const uint64_t ga_base  = (uint64_t)(uintptr_t)x + (uint64_t)(g0 * 3) * 4ull;
  const int nchunks = (S_LEN + CHUNK - 1) / CHUNK;

  // Issue this wave's slice of chunk c (rows [c*CHUNK + wave*RPW, +RPW) clipped to S).
  auto issue = [&](int c) -> bool {
    int t0 = c * CHUNK + wave * RPW;
    int rows = S_LEN - t0;
    rows = rows < RPW ? rows : RPW;
    if (rows > 0) {
      uint32_t off = lds_base + (uint32_t)(((c & 1) * CHUNK + wave * RPW) * GB3) * 4u;
      tdm_issue(ga_base + (uint64_t)t0 * rowbytes, off, rows);
      return true;
    }
    return false;
  };

  issue(0);  // prologue: chunk 0 into buffer 0

  for (int c = 0; c < nchunks; ++c) {
    bool more = false;
    if (c + 1 < nchunks) more = issue(c + 1);       // prefetch next chunk (other buffer)
    if (more) __builtin_amdgcn_s_wait_tensorcnt(1); // oldest outstanding (chunk c) done
    else      __builtin_amdgcn_s_wait_tensorcnt(0);
    __syncthreads();

    const float* xb = &xs[(c & 1) * CHUNK * GB3];
    int rows = S_LEN - c * CHUNK;
    rows = rows < CHUNK ? rows : CHUNK;
    for (int r = 0; r < rows; ++r) {
      const float p   = xb[r * GB3 + pl * 3 + 0];
      const float tC  = xb[r * GB3 + pl * 3 + 1];
      const float pet = xb[r * GB3 + pl * 3 + 2];

      // ---- HBV step ----
      float rain = (tC >= ttp) ? p : 0.0f;
      sp += (tC < ttp) ? p : 0.0f;
      float melt = fminf(fmaxf(cfmax * (tC - ttp), 0.0f), sp);
      mw += melt; sp -= melt;
      float refr = fminf(fmaxf(cfr * cfmax * (ttp - tC), 0.0f), mw);
      sp += refr; mw -= refr;
      float tosoil = fmaxf(mw - cwh * sp, 0.0f);
      mw -= tosoil;
      float sw = fminf(fmaxf(__powf(sm * inv_fc, beta), 0.0f), 1.0f);
      float rech = (rain + tosoil) * sw;
      sm += rain + tosoil - rech;
      float excess = fmaxf(sm - fc, 0.0f);
      sm -= excess;
      float et = fminf(pet * fminf(fmaxf(sm * inv_lpfc, 0.0f), 1.0f), sm);
      sm = fmaxf(sm - et, NEARZ);
      suz += rech + excess;
      float pa = fminf(perc, suz);
      suz -= pa;
      float q0 = k0 * fmaxf(suz - uzl, 0.0f);
      suz -= q0;
      float q1 = k1 * suz;
      suz -= q1;
      slz += pa;
      float q2 = k2 * slz;
      slz -= q2;
      float qh = q0 + q1 + q2;

      // ---- ExpHydro step (independent chain: interleaves with HBV for ILP) ----
      float ps = (tC <= tmin) ? p : 0.0f;
      float pr = p - ps;
      float meltx = (tC > tmax) ? fminf(s0, df * (tC - tmax)) : 0.0f;
      s0 += ps - meltx;
      s1 += pr + meltx;
      float qspill = fmaxf(s1 - smax, 0.0f); s1 -= qspill;
      float etx = fminf(pet * fminf(fmaxf(s1 * inv_smax, 0.0f), 1.0f), s1); s1 -= etx;
      float qb = fminf(qmax * __expf(-f * fmaxf(smax - s1, 0.0f)), s1); s1 -= qb;
      float qe = qspill + qb;

      // qsim = 0.5*(mean_m qh + mean_m qe) = 0.25 * sum over m of (qh+qe)
      float v = qh + qe;
      float vp = __shfl_xor(v, 1, 32);   // partner NMUL lane
      if (m == 0) qsim[(c * CHUNK + r) * G_N + g] = 0.25f * (v + vp);
    }
    __syncthreads();  // buffer (c&1) free for reuse at issue(c+2)
  }
}

// Kernel 2: gamma unit-hydrograph weights, one thread per gauge.
extern "C" __global__ void __launch_bounds__(256)
uh_gamma_kernel(const float* __restrict__ raw, float* __restrict__ w) {
  int g = blockIdx.x * 256 + threadIdx.x;
  if (g >= G_N) return;
  float a  = raw[g * RAWSTR + 36] * 2.9f;
  float b  = raw[g * RAWSTR + 37] * 6.5f;
  float aa = fmaxf(a, 0.0f) + 0.1f;
  float th = fmaxf(b, 0.0f) + 0.5f;
  float lg = lgammaf(aa);
  float lth = __logf(th);
  float inv_th = 1.0f / th;
  float wv[LENF];
  float sum = 0.0f;
#pragma unroll
  for (int k = 0; k < LENF; ++k) {
    float tk = (float)k + 0.5f;
    float logw = -lg - aa * lth + (aa - 1.0f) * __logf(tk) - tk * inv_th;
    wv[k] = __expf(logw);
    sum += wv[k];
  }
  float inv = 1.0f / sum;
#pragma unroll
  for (int k = 0; k < LENF; ++k) w[k * G_N + g] = wv[k] * inv;
}

// Kernel 3: causal 15-tap routing convolution. Thread per (t, g); fully coalesced;
// w and the 15-row qsim window are L2-resident (whole problem < 192 MB L2).
extern "C" __global__ void __launch_bounds__(256)
route_kernel(const float* __restrict__ qsim, const float* __restrict__ w,
             float* __restrict__ out) {
  int g = blockIdx.x * 256 + threadIdx.x;
  int t = blockIdx.y;
  if (g >= G_N) return;
  float acc = 0.0f;
#pragma unroll
  for (int k = 0; k < LENF; ++k) {
    if (k <= t) acc = fmaf(w[k * G_N + g], qsim[(t - k) * G_N + g], acc);
  }
  out[t * G_N + g] = acc;
}

extern "C" void kernel_launch(void* const* d_in, const int* in_sizes, int n_in,
                              void* d_out, int out_size, void* d_ws, size_t ws_size,
                              hipStream_t stream) {
  const float* x   = (const float*)d_in[0];   // (S, G, 3) f32
  const float* raw = (const float*)d_in[1];   // (G, 38) f32
  float* out  = (float*)d_out;                // (S, G) f32
  float* qsim = (float*)d_ws;                 // S*G floats
  float* w    = qsim + (size_t)S_LEN * G_N;   // LENF*G floats

  hydro_scan_kernel<<<dim3(G_N / GB), dim3(NTHR), 0, stream>>>(x, raw, qsim);
  uh_gamma_kernel<<<dim3((G_N + 255) / 256), dim3(256), 0, stream>>>(raw, w);
  route_kernel<<<dim3((G_N + 255) / 256, S_LEN), dim3(256), 0, stream>>>(qsim, w, out);
}